// GNNLayer_21706764714012
// MI455X (gfx1250) — compile-verified
//
#include <hip/hip_runtime.h>

#define F 128  // IN_F == OUT_F

typedef __attribute__((ext_vector_type(2))) float v2f;
typedef __attribute__((ext_vector_type(4))) float v4f;
typedef __attribute__((ext_vector_type(8))) float v8f;

// ---------------------------------------------------------------------------
// Phase 0: zero the 4 aggregate buffers in workspace (must be done every call;
// harness poisons d_ws once and never re-zeroes).
// ---------------------------------------------------------------------------
__global__ void zero_ws_kernel(float* __restrict__ p, long long n_floats) {
    long long i = (long long)blockIdx.x * blockDim.x + threadIdx.x;
    const long long stride = (long long)gridDim.x * blockDim.x;
    const long long n4 = n_floats >> 2;
    v4f z = {0.f, 0.f, 0.f, 0.f};
    for (; i < n4; i += stride)
        ((v4f*)p)[i] = z;
}

// ---------------------------------------------------------------------------
// Phase 1: fused dual SpMM. For each COO edge (r,c,v):
//   aggF[r, :] += v * X[c, :]
//   aggI[r, :] += v * X[c, :]^2      (inter_feature fused, never materialized)
// One wave per edge-slot; 32 lanes cover a 32-column feature chunk so the
// atomic destination working set (102 MB/pass) stays resident in the 192 MB L2.
// Edge streams use NT loads + prefetch so they don't evict the hot agg lines.
// ---------------------------------------------------------------------------
__global__ void __launch_bounds__(256)
spmm_dual_kernel(const int* __restrict__ rows, const int* __restrict__ cols,
                 const float* __restrict__ vals, const float* __restrict__ X,
                 float* __restrict__ aggF, float* __restrict__ aggI,
                 int nnz, int colOff) {
    const int lane = threadIdx.x;                       // 0..31 -> feature within chunk
    int e = blockIdx.x * blockDim.y + threadIdx.y;
    const int estride = gridDim.x * blockDim.y;
    for (; e < nnz; e += estride) {
        __builtin_prefetch(rows + e + estride, 0, 1);   // global_prefetch_b8
        __builtin_prefetch(cols + e + estride, 0, 1);
        __builtin_prefetch(vals + e + estride, 0, 1);
        const int   r = __builtin_nontemporal_load(rows + e);
        const int   c = __builtin_nontemporal_load(cols + e);
        const float v = __builtin_nontemporal_load(vals + e);
        const float x = X[(long long)c * F + colOff + lane];   // RT: L2-resident
        const float m = v * x;
        const long long o = (long long)r * F + colOff + lane;
        unsafeAtomicAdd(aggF + o, m);        // global_atomic_add_f32 (no-return)
        unsafeAtomicAdd(aggI + o, m * x);
    }
}

// ---------------------------------------------------------------------------
// Phase 2: fused GEMM via V_WMMA_F32_16X16X4_F32.
//   out = agg0@W0 + agg1@W1 + agg2@W2 + agg3@W3 + (b0+b1+b2+b3)
// K folded to 512 across the 4 stacked aggregates. One wave owns a 16x16
// output tile; a 256-thread block (8 waves) covers all 8 column tiles of a
// 16-row stripe. f32 WMMA keeps reference precision; phase is HBM-bound so
// the x4 f32 matrix rate is not the limiter.
//
// ISA VGPR layouts (CDNA5 ISA):
//   A 16x4 f32 : lane l(+16h) -> A[M=l][k+2h], A[M=l][k+2h+1]   (v2f)
//   B 4x16 f32 : lane l(+16h) -> B[k+2h][n=l], B[k+2h+1][n=l]   (v2f)
//   C/D 16x16  : VGPR j, lane l(+16h) -> D[M = j + 8h][N = l]
// ---------------------------------------------------------------------------
__global__ void __launch_bounds__(256)
wmma_gemm_kernel(const float* __restrict__ agg, long long perBuf,
                 const float* __restrict__ W0, const float* __restrict__ W1,
                 const float* __restrict__ W2, const float* __restrict__ W3,
                 const float* __restrict__ b0, const float* __restrict__ b1,
                 const float* __restrict__ b2, const float* __restrict__ b3,
                 float* __restrict__ out, int N) {
    const int wave = threadIdx.x >> 5;      // 0..7 -> output column tile
    const int lane = threadIdx.x & 31;
    const int half = lane >> 4;             // 0 | 1
    const int l    = lane & 15;
    const int rowBase = blockIdx.x << 4;
    const int nBase   = wave << 4;

    const float* Ws[4] = {W0, W1, W2, W3};

    // clamp for a (non-occurring here: N % 16 == 0) tail stripe; WMMA needs
    // EXEC all-ones so we clamp addresses instead of predicating lanes
    const int rA = (rowBase + l < N) ? (rowBase + l) : (N - 1);

    v8f c = {0.f, 0.f, 0.f, 0.f, 0.f, 0.f, 0.f, 0.f};
#pragma unroll
    for (int b = 0; b < 4; ++b) {
        const float* A = agg + (long long)b * perBuf + (long long)rA * F;
        const float* W = Ws[b] + nBase + l;
#pragma unroll 8
        for (int k = 0; k < F; k += 4) {
            const int ka = k + (half << 1);
            v2f av = *(const v2f*)(A + ka);            // global_load_b64
            v2f bv;
            bv.x = W[ka * F];
            bv.y = W[(ka + 1) * F];
            c = __builtin_amdgcn_wmma_f32_16x16x4_f32(
                    false, av, false, bv, (short)0, c, false, false);
        }
    }

    const float bs = b0[nBase + l] + b1[nBase + l] + b2[nBase + l] + b3[nBase + l];
#pragma unroll
    for (int j = 0; j < 8; ++j) {
        const int rr = rowBase + j + (half << 3);
        if (rr < N)
            __builtin_nontemporal_store(c[j] + bs, out + (long long)rr * F + nBase + l);
    }
}

// ---------------------------------------------------------------------------
// Launch
// ---------------------------------------------------------------------------
extern "C" void kernel_launch(void* const* d_in, const int* in_sizes, int n_in,
                              void* d_out, int out_size, void* d_ws, size_t ws_size,
                              hipStream_t stream) {
    const float* X      = (const float*)d_in[0];
    const int*   Lr     = (const int*)  d_in[1];
    const int*   Lc     = (const int*)  d_in[2];
    const float* Lv     = (const float*)d_in[3];
    const int*   Ur     = (const int*)  d_in[4];
    const int*   Uc     = (const int*)  d_in[5];
    const float* Uv     = (const float*)d_in[6];
    const float* W_lin  = (const float*)d_in[7];
    const float* b_lin  = (const float*)d_in[8];
    const float* W_lin1 = (const float*)d_in[9];
    const float* b_lin1 = (const float*)d_in[10];
    const float* W_iat  = (const float*)d_in[11];
    const float* b_iat  = (const float*)d_in[12];
    const float* W_iat1 = (const float*)d_in[13];
    const float* b_iat1 = (const float*)d_in[14];

    const int N   = in_sizes[0] / F;
    const int nnz = in_sizes[1];

    float* agg = (float*)d_ws;                 // 4 x [N, F] f32 aggregates
    const long long perBuf = (long long)N * F;

    zero_ws_kernel<<<4096, 256, 0, stream>>>(agg, 4 * perBuf);

    // buffer0 = L@X   -> W_lin   ; buffer1 = L@X^2  -> W_iat
    // buffer2 = UI@X  -> W_lin1  ; buffer3 = UI@X^2 -> W_iat1
    dim3 sblk(32, 8);
    const int sgrid = 8192;
    for (int co = 0; co < F; co += 32) {       // 4 L2-resident column passes
        spmm_dual_kernel<<<sgrid, sblk, 0, stream>>>(
            Lr, Lc, Lv, X, agg, agg + perBuf, nnz, co);
        spmm_dual_kernel<<<sgrid, sblk, 0, stream>>>(
            Ur, Uc, Uv, X, agg + 2 * perBuf, agg + 3 * perBuf, nnz, co);
    }

    const int rowTiles = (N + 15) / 16;
    wmma_gemm_kernel<<<rowTiles, 256, 0, stream>>>(
        agg, perBuf, W_lin, W_iat, W_lin1, W_iat1,
        b_lin, b_iat, b_lin1, b_iat1, (float*)d_out, N);
}